// ProductRepresentation_29291676959152
// MI455X (gfx1250) — compile-verified
//
#include <hip/hip_runtime.h>
#include <math.h>

#define DIM   1024
#define NPAR  (DIM*(DIM-1)/2)     // 523776 rotations
#define MSZ   (DIM*DIM)           // floats per matrix (4 MB)

typedef float v2f __attribute__((ext_vector_type(2)));
typedef float v8f __attribute__((ext_vector_type(8)));

// ---------------- Phase 0: cos/sin table ----------------
__global__ __launch_bounds__(256) void trig_kernel(const float* __restrict__ th,
                                                   float* __restrict__ cs, int n) {
  int k = blockIdx.x * 256 + threadIdx.x;
  if (k < n) {
    float t = th[k];
    cs[2*k]     = cosf(t);
    cs[2*k + 1] = sinf(t);
  }
}

// ---------------- Phase 1: build chunk matrices ----------------
// grid.x = C*32 blocks, 32 threads/block (1 wave), dynamic LDS = 32*1025*4 B.
// Each lane owns one row; stride 1025 => 32 lanes on one column hit 32
// distinct banks. Inner loops are segmented per rotation-group so the hot
// loop (ds_load + 2 fma + ds_store) has no boundary branch.
__global__ __launch_bounds__(32) void chunk_kernel(const float* __restrict__ thetas,
                                                   const float* __restrict__ cs,
                                                   int useCS,
                                                   float* __restrict__ outBase,
                                                   int kper) {
  extern __shared__ float S[];
  const int lane  = threadIdx.x;
  const int chunk = blockIdx.x >> 5;
  const int rblk  = blockIdx.x & 31;
  const int grow  = rblk * 32 + lane;
  const int base  = lane * 1025;

  for (int col = 0; col < DIM; ++col)
    S[base + col] = (col == grow) ? 1.0f : 0.0f;

  const int k0 = chunk * kper, k1 = k0 + kper;
  // locate (i,p) of rotation k0: offset(i) = sum_{t<i}(DIM-1-t)
  int i = 0, off = 0;
  while (off + (DIM - 1 - i) <= k0) { off += (DIM - 1 - i); ++i; }
  int p = i + 1 + (k0 - off);

  const float2* cs2 = (const float2*)cs;
  int k = k0;
  while (k < k1) {
    int kend = k + (DIM - p);           // rotations left in group i
    if (kend > k1) kend = k1;
    float xi = S[base + i];
    if (useCS) {
      for (; k < kend; ++k, ++p) {
        float2 w  = cs2[k];             // uniform across lanes
        float  xp = S[base + p];
        float nxi = fmaf(w.x, xi, -w.y * xp);
        S[base + p] = fmaf(w.y, xi, w.x * xp);
        xi = nxi;
      }
    } else {                            // fallback: no room for c/s table
      for (; k < kend; ++k, ++p) {
        float t = thetas[k];
        float c = cosf(t), s = sinf(t);
        float xp  = S[base + p];
        float nxi = fmaf(c, xi, -s * xp);
        S[base + p] = fmaf(s, xi, c * xp);
        xi = nxi;
      }
    }
    S[base + i] = xi;
    if (p == DIM) { ++i; p = i + 1; }
  }

  __syncthreads();                      // lanes read each other's rows below

  for (int r = 0; r < 32; ++r) {
    int row = rblk * 32 + r;
    float* dst = outBase + (size_t)chunk * MSZ + (size_t)row * DIM;
    for (int col = lane; col < DIM; col += 32)
      dst[col] = S[r * 1025 + col];     // coalesced global store
  }
}

// ---------------- Phase 2: batched fp32 GEMM via V_WMMA_F32_16X16X4_F32 ----
// 128 threads = 4 waves; block tile 64x64; K-tile 32 (8 k-steps of 4).
// Fragments pre-paired in LDS as v2f: AsP[s][h*80+m] = {A[m][4s+2h],
// A[m][4s+2h+1]}. h-stride 80 v2f = 640 B = +32 banks, so lanes 0-15 (h=0)
// and 16-31 (h=1) read disjoint banks: one conflict-free ds_load_b64 per
// fragment, zero shuffle movs in front of the WMMAs.
__global__ __launch_bounds__(128) void gemm_kernel(const float* __restrict__ Ab,
                                                   const float* __restrict__ Bb,
                                                   float* __restrict__ Cb,
                                                   long long sA, long long sB,
                                                   long long sC) {
  __shared__ v2f AsP[8][160];   // [kstep][h*80 + m]
  __shared__ v2f BsP[8][160];   // [kstep][h*80 + n]

  const float* A = Ab + (long long)blockIdx.z * sA;
  const float* B = Bb + (long long)blockIdx.z * sB;
  float*       C = Cb + (long long)blockIdx.z * sC;

  const int tid  = threadIdx.x;
  const int lane = tid & 31;
  const int wave = tid >> 5;
  const int wm   = wave >> 1, wn = wave & 1;
  const int lrow = lane & 15;
  const int h    = lane >> 4;           // 0: K+0/1, 1: K+2/3

  const int bm = blockIdx.y * 64;
  const int bn = blockIdx.x * 64;

  v8f acc00 = {}, acc01 = {}, acc10 = {}, acc11 = {};

  const int lmA = tid >> 3;             // 0..15: m group for A staging
  const int sAk = tid & 7;              // k-step for A staging
  const int n4  = (tid & 15) << 2;      // 0..60: n quad for B staging
  const int sBk = tid >> 4;             // 0..7: k-step for B staging

  for (int kt = 0; kt < DIM; kt += 32) {
#pragma unroll
    for (int it = 0; it < 4; ++it) {    // A: 64m x 32k, transposed+paired
      int m = lmA + it * 16;
      float4 va = *(const float4*)(A + (size_t)(bm + m) * DIM + kt + sAk * 4);
      v2f lo; lo.x = va.x; lo.y = va.y;
      v2f hi; hi.x = va.z; hi.y = va.w;
      AsP[sAk][m]      = lo;
      AsP[sAk][80 + m] = hi;
    }
    {                                   // B: 32k x 64n, k paired per n
      const float* Bp = B + (size_t)(kt + sBk * 4) * DIM + bn + n4;
      float4 r0 = *(const float4*)(Bp);
      float4 r1 = *(const float4*)(Bp + DIM);
      float4 r2 = *(const float4*)(Bp + 2 * DIM);
      float4 r3 = *(const float4*)(Bp + 3 * DIM);
      v2f t;
      t.x = r0.x; t.y = r1.x; BsP[sBk][n4 + 0]      = t;
      t.x = r0.y; t.y = r1.y; BsP[sBk][n4 + 1]      = t;
      t.x = r0.z; t.y = r1.z; BsP[sBk][n4 + 2]      = t;
      t.x = r0.w; t.y = r1.w; BsP[sBk][n4 + 3]      = t;
      t.x = r2.x; t.y = r3.x; BsP[sBk][80 + n4 + 0] = t;
      t.x = r2.y; t.y = r3.y; BsP[sBk][80 + n4 + 1] = t;
      t.x = r2.z; t.y = r3.z; BsP[sBk][80 + n4 + 2] = t;
      t.x = r2.w; t.y = r3.w; BsP[sBk][80 + n4 + 3] = t;
    }
    __syncthreads();
#pragma unroll
    for (int s = 0; s < 8; ++s) {
      v2f a0 = AsP[s][h * 80 + wm * 32 + lrow];
      v2f a1 = AsP[s][h * 80 + wm * 32 + 16 + lrow];
      v2f b0 = BsP[s][h * 80 + wn * 32 + lrow];
      v2f b1 = BsP[s][h * 80 + wn * 32 + 16 + lrow];
      acc00 = __builtin_amdgcn_wmma_f32_16x16x4_f32(false, a0, false, b0, (short)0, acc00, false, false);
      acc01 = __builtin_amdgcn_wmma_f32_16x16x4_f32(false, a0, false, b1, (short)0, acc01, false, false);
      acc10 = __builtin_amdgcn_wmma_f32_16x16x4_f32(false, a1, false, b0, (short)0, acc10, false, false);
      acc11 = __builtin_amdgcn_wmma_f32_16x16x4_f32(false, a1, false, b1, (short)0, acc11, false, false);
    }
    __syncthreads();
  }

  const int mo   = h << 3;              // lanes 16-31 hold M=8..15
  const int colb = bn + wn * 32 + lrow;
#pragma unroll
  for (int v = 0; v < 8; ++v) {
    C[(size_t)(bm + wm*32 +      mo + v) * DIM + colb     ] = acc00[v];
    C[(size_t)(bm + wm*32 +      mo + v) * DIM + colb + 16] = acc01[v];
    C[(size_t)(bm + wm*32 + 16 + mo + v) * DIM + colb     ] = acc10[v];
    C[(size_t)(bm + wm*32 + 16 + mo + v) * DIM + colb + 16] = acc11[v];
  }
}

// ---------------- Host orchestration ----------------
extern "C" void kernel_launch(void* const* d_in, const int* in_sizes, int n_in,
                              void* d_out, int out_size, void* d_ws, size_t ws_size,
                              hipStream_t stream) {
  (void)in_sizes; (void)n_in; (void)out_size;
  const float* thetas = (const float*)d_in[0];   // i_idx/p_idx derived analytically
  float* out = (float*)d_out;
  float* ws  = (float*)d_ws;

  // pick largest chunk count C whose workspace fits: (1 + C + C/2) matrices
  const size_t MB4 = (size_t)MSZ * sizeof(float);
  int C = 1;
  const int cands[5] = {32, 16, 8, 4, 2};
  for (int ci = 0; ci < 5; ++ci) {
    size_t need = MB4 * (size_t)(1 + cands[ci] + cands[ci] / 2);
    if (ws_size >= need) { C = cands[ci]; break; }
  }
  int useCS = (ws_size >= (size_t)(2 * NPAR) * sizeof(float)) ? 1 : 0;

  float* cs     = ws;                          // 2*NPAR floats (< 1 MSZ)
  float* arena0 = ws + (size_t)MSZ;            // C chunk matrices
  float* arena1 = ws + (size_t)MSZ * (1 + C);  // C/2 intermediates

  if (useCS)
    trig_kernel<<<(NPAR + 255) / 256, 256, 0, stream>>>(thetas, cs, NPAR);

  float* p1out = (C == 1) ? out : arena0;
  int kper = NPAR / C;                         // NPAR divisible by 2..32
  chunk_kernel<<<dim3(C * 32), dim3(32), 32 * 1025 * sizeof(float), stream>>>(
      thetas, cs, useCS, p1out, kper);

  if (C > 1) {                                 // binary-tree fold: M = A0*A1*...*A(C-1)
    float* src = arena0;
    float* spare = arena1;
    int cnt = C;
    while (cnt > 1) {
      int nb = cnt >> 1;
      float* dst = (nb == 1) ? out : spare;
      gemm_kernel<<<dim3(DIM / 64, DIM / 64, nb), dim3(128), 0, stream>>>(
          src, src + MSZ, dst,
          (long long)2 * MSZ, (long long)2 * MSZ, (long long)MSZ);
      spare = src;
      src = dst;
      cnt = nb;
    }
  }
}